// GraphConvolution_62105227100574
// MI455X (gfx1250) — compile-verified
//
#include <hip/hip_runtime.h>
#include <hip/hip_bf16.h>

typedef __attribute__((ext_vector_type(2))) float v2f;
typedef __attribute__((ext_vector_type(8))) float v8f;

#define DIM 256

// ---------------------------------------------------------------------------
// Kernel 1: Y = X @ W   (fp32 WMMA, 16x16x4)
// Block = 256 threads = 8 waves. Block computes rows [16*bid, 16*bid+16),
// wave w computes columns [32*w, 32*w+32) as two 16x16 WMMA tiles.
// ---------------------------------------------------------------------------
__global__ void __launch_bounds__(256)
gemm_xw_kernel(const float* __restrict__ X, const float* __restrict__ W,
               float* __restrict__ Y) {
  const int wave = threadIdx.x >> 5;
  const int lane = threadIdx.x & 31;
  const int half = lane >> 4;    // 0: lanes 0-15, 1: lanes 16-31
  const int l15  = lane & 15;

  const int m0 = blockIdx.x * 16;
  const int n0 = wave * 32;

  const float* __restrict__ Arow = X + (size_t)(m0 + l15) * DIM;

  v8f acc0 = {};
  v8f acc1 = {};

  const int col0 = n0 + l15;
  const int col1 = n0 + 16 + l15;

  for (int k = 0; k < DIM; k += 4) {
    // A fragment (16x4 f32): lane holds row M=l15; VGPR0 = K=k+2*half,
    // VGPR1 = K=k+2*half+1  -> contiguous pair in the row.
    const int ka = k + 2 * half;
    v2f a;
    a.x = Arow[ka];
    a.y = Arow[ka + 1];

    // B fragments (4x16 f32): lane holds col N; VGPR0 = K=ka, VGPR1 = K=ka+1.
    v2f b0, b1;
    b0.x = W[(size_t)ka * DIM + col0];
    b0.y = W[(size_t)(ka + 1) * DIM + col0];
    b1.x = W[(size_t)ka * DIM + col1];
    b1.y = W[(size_t)(ka + 1) * DIM + col1];

    acc0 = __builtin_amdgcn_wmma_f32_16x16x4_f32(
        /*neg_a=*/false, a, /*neg_b=*/false, b0,
        /*c_mod=*/(short)0, acc0, /*reuse_a=*/false, /*reuse_b=*/false);
    acc1 = __builtin_amdgcn_wmma_f32_16x16x4_f32(
        /*neg_a=*/false, a, /*neg_b=*/false, b1,
        /*c_mod=*/(short)0, acc1, /*reuse_a=*/false, /*reuse_b=*/false);
  }

  // C/D layout: element (M = v + 8*half, N = l15) in VGPR v.
  #pragma unroll
  for (int v = 0; v < 8; ++v) {
    const size_t row = (size_t)(m0 + v + 8 * half) * DIM;
    Y[row + col0] = acc0[v];
    Y[row + col1] = acc1[v];
  }
}

// ---------------------------------------------------------------------------
// Kernel 2: out[n, j] = bias[j]   (base for the scatter-add; covers rows
// with no incoming edges)
// ---------------------------------------------------------------------------
__global__ void __launch_bounds__(256)
init_bias_kernel(float* __restrict__ Z, const float* __restrict__ bias,
                 int total) {
  const int idx = blockIdx.x * 256 + threadIdx.x;
  if (idx < total) {
    Z[idx] = bias[idx & (DIM - 1)];
  }
}

// ---------------------------------------------------------------------------
// Kernel 3: out[row[e]] += val[e] * Y[col[e]]  -- one wave per edge.
// 32 lanes x float4 = 128 floats/iter, 2 iters covers 256 columns.
// fp32 HW atomics (relaxed, device scope) resolve in L2 (Y + out are
// L2-resident: 100MB each vs 192MB L2).
// ---------------------------------------------------------------------------
__global__ void __launch_bounds__(256)
spmm_scatter_kernel(const int* __restrict__ erow, const int* __restrict__ ecol,
                    const float* __restrict__ eval,
                    const float* __restrict__ Y, float* __restrict__ Z,
                    int nedges) {
  const int gwarp = (int)((blockIdx.x * 256 + threadIdx.x) >> 5);
  const int lane = threadIdx.x & 31;
  if (gwarp >= nedges) return;

  const int r = erow[gwarp];
  const int c = ecol[gwarp];
  const float v = eval[gwarp];

  const float4* __restrict__ ysrc =
      (const float4*)(Y + (size_t)c * DIM);
  float* __restrict__ zdst = Z + (size_t)r * DIM;

  #pragma unroll
  for (int i = 0; i < 2; ++i) {
    const int q = lane + 32 * i;          // float4 index within the row
    const float4 y = ysrc[q];
    const int base = q * 4;
    __hip_atomic_fetch_add(zdst + base + 0, v * y.x,
                           __ATOMIC_RELAXED, __HIP_MEMORY_SCOPE_AGENT);
    __hip_atomic_fetch_add(zdst + base + 1, v * y.y,
                           __ATOMIC_RELAXED, __HIP_MEMORY_SCOPE_AGENT);
    __hip_atomic_fetch_add(zdst + base + 2, v * y.z,
                           __ATOMIC_RELAXED, __HIP_MEMORY_SCOPE_AGENT);
    __hip_atomic_fetch_add(zdst + base + 3, v * y.w,
                           __ATOMIC_RELAXED, __HIP_MEMORY_SCOPE_AGENT);
  }
}

// ---------------------------------------------------------------------------
// launch:  out = A @ (X @ W) + b   ==  (A @ X) @ W + b
// inputs: X[0], edge_row[1], edge_col[2], edge_vals[3], weight[4], bias[5]
// ---------------------------------------------------------------------------
extern "C" void kernel_launch(void* const* d_in, const int* in_sizes, int n_in,
                              void* d_out, int out_size, void* d_ws, size_t ws_size,
                              hipStream_t stream) {
  const float* X    = (const float*)d_in[0];
  const int*   erow = (const int*)d_in[1];
  const int*   ecol = (const int*)d_in[2];
  const float* evl  = (const float*)d_in[3];
  const float* W    = (const float*)d_in[4];
  const float* bias = (const float*)d_in[5];
  float* Z = (float*)d_out;

  const int n_nodes = in_sizes[0] / DIM;   // 100000
  const int n_edges = in_sizes[1];         // 1600000

  float* Y = (float*)d_ws;                 // [n_nodes, 256] fp32 scratch

  // 1) Y = X @ W  (WMMA)
  gemm_xw_kernel<<<n_nodes / 16, 256, 0, stream>>>(X, W, Y);

  // 2) Z = broadcast(bias)
  const int total = n_nodes * DIM;
  init_bias_kernel<<<(total + 255) / 256, 256, 0, stream>>>(Z, bias, total);

  // 3) Z += A @ Y   (one wave per edge)
  const int warps_per_block = 256 / 32;
  const int nblocks = (n_edges + warps_per_block - 1) / warps_per_block;
  spmm_scatter_kernel<<<nblocks, 256, 0, stream>>>(erow, ecol, evl, Y, Z, n_edges);
}